// LIFSpike_81312320847922
// MI455X (gfx1250) — compile-verified
//
#include <hip/hip_runtime.h>

// LIF spiking-neuron forward (T=4 scan over leaky-integrate-fire), fp32.
// Memory-bound streaming kernel: 128-bit non-temporal loads/stores, one pass.

typedef __attribute__((ext_vector_type(4))) float v4f;

#define LIF_T 4
__constant__ const float kTau = 0.5f;
__constant__ const float kThresh = 1.0f;

__global__ __launch_bounds__(256) void lif_forward_v4(
    const float* __restrict__ x, float* __restrict__ out, unsigned int n4) {
  // n4 = number of float4 chunks per timestep. Per-timestep stride = n4*4 floats.
  unsigned int i = blockIdx.x * blockDim.x + threadIdx.x;
  if (i >= n4) return;

  const unsigned long long base   = (unsigned long long)i * 4ull;
  const unsigned long long stride = (unsigned long long)n4 * 4ull;

  // Issue all 4 timestep loads up front (4 global_load_b128 in flight, NT hint).
  v4f xv[LIF_T];
#pragma unroll
  for (int t = 0; t < LIF_T; ++t) {
    const v4f* p = (const v4f*)(x + base + (unsigned long long)t * stride);
    xv[t] = __builtin_nontemporal_load(p);
  }

  float mem0 = 0.f, mem1 = 0.f, mem2 = 0.f, mem3 = 0.f;
#pragma unroll
  for (int t = 0; t < LIF_T; ++t) {
    v4f s;
    // component 0
    mem0 = mem0 * 0.5f + xv[t].x;
    { float sp = (mem0 >= 1.0f) ? 1.0f : 0.0f; mem0 -= sp; s.x = sp; }
    // component 1
    mem1 = mem1 * 0.5f + xv[t].y;
    { float sp = (mem1 >= 1.0f) ? 1.0f : 0.0f; mem1 -= sp; s.y = sp; }
    // component 2
    mem2 = mem2 * 0.5f + xv[t].z;
    { float sp = (mem2 >= 1.0f) ? 1.0f : 0.0f; mem2 -= sp; s.z = sp; }
    // component 3
    mem3 = mem3 * 0.5f + xv[t].w;
    { float sp = (mem3 >= 1.0f) ? 1.0f : 0.0f; mem3 -= sp; s.w = sp; }

    v4f* po = (v4f*)(out + base + (unsigned long long)t * stride);
    __builtin_nontemporal_store(s, po);
  }
}

// Scalar tail for n_spatial not divisible by 4 (not hit for the reference shape).
__global__ __launch_bounds__(256) void lif_forward_tail(
    const float* __restrict__ x, float* __restrict__ out,
    unsigned int start, unsigned int n_spatial) {
  unsigned int i = start + blockIdx.x * blockDim.x + threadIdx.x;
  if (i >= n_spatial) return;
  float mem = 0.f;
#pragma unroll
  for (int t = 0; t < LIF_T; ++t) {
    unsigned long long idx = (unsigned long long)t * n_spatial + i;
    mem = mem * 0.5f + x[idx];
    float sp = (mem >= 1.0f) ? 1.0f : 0.0f;
    mem -= sp;
    out[idx] = sp;
  }
}

extern "C" void kernel_launch(void* const* d_in, const int* in_sizes, int n_in,
                              void* d_out, int out_size, void* d_ws, size_t ws_size,
                              hipStream_t stream) {
  (void)n_in; (void)out_size; (void)d_ws; (void)ws_size;
  const float* x = (const float*)d_in[0];
  float* out = (float*)d_out;

  const unsigned int total     = (unsigned int)in_sizes[0];  // T * N_spatial
  const unsigned int n_spatial = total / LIF_T;              // 8,388,608
  const unsigned int n4        = n_spatial / 4;              // float4 chunks
  const unsigned int rem       = n_spatial - n4 * 4;

  if (n4 > 0) {
    dim3 block(256);
    dim3 grid((n4 + 255u) / 256u);
    lif_forward_v4<<<grid, block, 0, stream>>>(x, out, n4);
  }
  if (rem > 0) {
    dim3 block(256);
    dim3 grid((rem + 255u) / 256u);
    lif_forward_tail<<<grid, block, 0, stream>>>(x, out, n4 * 4, n_spatial);
  }
}